// WaveFDTD2D_58926951301514
// MI455X (gfx1250) — compile-verified
//
#include <hip/hip_runtime.h>

// ---------------- problem constants (match reference) ----------------
#define NXg     512
#define NZg     512
#define NSTEPSg 512
#define NRECg   128
#define Ncells  (NXg * NZg)

__device__ __constant__ float kDT2      = 1.0e-6f;   // (1e-3)^2
__device__ __constant__ float kINV_DXDZ = 1.0f / (10.0f * 10.0f);

// ---------------- persistent-kernel geometry ----------------
#define NB   32                    // persistent workgroups (must all be co-resident)
#define TPB  1024                  // 32 wave32 waves per workgroup
#define ROWS (NXg / NB)            // 16 rows of the grid per block
#define TROWS (ROWS + 2)           // + top/bottom halo rows in LDS
#define CELLS_PER_THREAD (ROWS * NZg / TPB)   // 8

// ---------------- CDNA5 async global->LDS copy ----------------
#if __has_builtin(__builtin_amdgcn_global_load_async_to_lds_b128)
#define HAVE_ASYNC_LDS 1
#endif

typedef int v4i __attribute__((ext_vector_type(4)));
typedef __attribute__((address_space(1))) v4i gv4i;   // global
typedef __attribute__((address_space(3))) v4i lv4i;   // LDS

__device__ __forceinline__ void async_copy_b128(const float* gp, float* lp) {
#if defined(HAVE_ASYNC_LDS)
    // per-lane 16B: copies from the global address in gp to the LDS address in lp   (tracked by ASYNCcnt)
    __builtin_amdgcn_global_load_async_to_lds_b128((gv4i*)gp, (lv4i*)lp, 0, 0);
#else
#warning "gfx1250 async-to-LDS builtin unavailable; using synchronous LDS staging fallback"
    *(float4*)lp = *(const float4*)gp;
#endif
}

__device__ __forceinline__ void wait_async_done() {
#if __has_builtin(__builtin_amdgcn_s_wait_asynccnt)
    __builtin_amdgcn_s_wait_asynccnt(0);
#else
    asm volatile("s_wait_asynccnt 0" ::: "memory");
#endif
}

// ---------------- init: zero p(t=0) buffer and the barrier counter ----------------
__global__ void WaveFDTD2D_init_kernel(float* p0, unsigned* bar) {
    unsigned i = blockIdx.x * blockDim.x + threadIdx.x;
    if (i < (unsigned)Ncells) p0[i] = 0.0f;
    if (i == 0) *bar = 0u;
}

// ---------------- persistent FDTD kernel: all 512 steps in one launch ----------------
__global__ void __launch_bounds__(TPB)
WaveFDTD2D_58926951301514_kernel(const float* __restrict__ vel,
                                 const float* __restrict__ src,
                                 const int*   __restrict__ src_x,
                                 const int*   __restrict__ src_z,
                                 const int*   __restrict__ rec_x,
                                 const int*   __restrict__ rec_z,
                                 float*       __restrict__ out,     // [NREC, NSTEPS]
                                 float*                    buf,     // 2 * Ncells floats
                                 unsigned*                 bar) {
    __shared__ float tile[TROWS][NZg];          // 18*512*4 = 36 KB of the 320 KB/WGP LDS

    const int tid = threadIdx.x;
    const int b   = blockIdx.x;
    const int x0  = b * ROWS;

    float* bufA = buf;
    float* bufB = buf + Ncells;

    const int sidx = src_x[0] * NZg + src_z[0];

    // ---- time-invariant / cell-private state carried in registers ----
    // po_reg[k] = p(t-1) at this thread's k-th cell (p(-1) == 0)
    // v2[k]     = vel^2 * DT2 at this thread's k-th cell (read from HBM exactly once)
    float po_reg[CELLS_PER_THREAD];
    float v2[CELLS_PER_THREAD];
#pragma unroll
    for (int k = 0; k < CELLS_PER_THREAD; ++k) {
        const int c  = tid + k * TPB;
        const int g  = (x0 + (c >> 9)) * NZg + (c & (NZg - 1));
        const float v = vel[g];
        v2[k]     = v * v * kDT2;
        po_reg[k] = 0.0f;
    }

    for (int t = 0; t < NSTEPSg; ++t) {
        const float* p  = (t & 1) ? bufB : bufA;   // p(t)   — read via LDS staging
        float*       pn = (t & 1) ? bufA : bufB;   // p(t+1) — written (was p(t-1), now dead)

        // ---- stage p slab (rows x0-1 .. x0+ROWS) into LDS, 16B per lane, async ----
        for (int i = tid * 4; i < TROWS * NZg; i += TPB * 4) {
            const int r  = i >> 9;              // NZg == 512
            const int z  = i & (NZg - 1);
            const int gx = x0 - 1 + r;          // uniform per wave32 (tid>>7 uniform)
            float* lp = &tile[r][z];
            if (gx >= 0 && gx < NXg) {
                async_copy_b128(p + (size_t)gx * NZg + z, lp);
            } else {                             // zero-padded boundary halo
                lp[0] = 0.0f; lp[1] = 0.0f; lp[2] = 0.0f; lp[3] = 0.0f;
            }
        }
        wait_async_done();
        __syncthreads();

        const float st = src[t];

        // ---- 5-point stencil update: 8 cells/thread, LDS-fed, p_old from registers ----
#pragma unroll
        for (int k = 0; k < CELLS_PER_THREAD; ++k) {
            const int c  = tid + k * TPB;
            const int xl = c >> 9;
            const int z  = c & (NZg - 1);
            const int r  = xl + 1;
            const int g  = (x0 + xl) * NZg + z;

            const float pc  = tile[r][z];
            float lap = tile[r - 1][z] + tile[r + 1][z]
                      + ((z > 0)       ? tile[r][z - 1] : 0.0f)
                      + ((z < NZg - 1) ? tile[r][z + 1] : 0.0f)
                      - 4.0f * pc;
            lap *= kINV_DXDZ;
            float val = 2.0f * pc - po_reg[k] + v2[k] * lap;
            if (g == sidx) val += st * kDT2;     // source injection
            pn[g]     = val;
            po_reg[k] = pc;                       // p(t) becomes p_old for step t+1
        }

        // ---- device-wide step barrier (32 co-resident blocks) ----
        __syncthreads();                          // tile reads done; block's stores issued
        if (tid == 0) {
            __threadfence();                      // release: flush to device scope (L2)
            atomicAdd(bar, 1u);
            const unsigned target = (unsigned)(t + 1) * NB;
            while (__hip_atomic_load(bar, __ATOMIC_RELAXED, __HIP_MEMORY_SCOPE_AGENT) < target)
                __builtin_amdgcn_s_sleep(1);
            __threadfence();                      // acquire: invalidate stale WGP$ lines
        }
        __syncthreads();

        // ---- record receivers for step t (pn is read-only during step t+1) ----
        if (b == 0 && tid < NRECg) {
            const int rg = rec_x[tid] * NZg + rec_z[tid];
            out[tid * NSTEPSg + t] = pn[rg];
        }
    }
}

// ---------------- host launch ----------------
extern "C" void kernel_launch(void* const* d_in, const int* in_sizes, int n_in,
                              void* d_out, int out_size, void* d_ws, size_t ws_size,
                              hipStream_t stream) {
    const float* vel   = (const float*)d_in[0];   // [512*512]
    const float* src   = (const float*)d_in[1];   // [512]
    const int*   src_x = (const int*)  d_in[2];   // [1]
    const int*   src_z = (const int*)  d_in[3];   // [1]
    const int*   rec_x = (const int*)  d_in[4];   // [128]
    const int*   rec_z = (const int*)  d_in[5];   // [128]
    float*       out   = (float*)d_out;           // [128*512]

    // workspace layout: [0,256): barrier counter; [256, 256+2*Ncells*4): p ping-pong buffers
    unsigned* bar = (unsigned*)d_ws;
    float*    buf = (float*)((char*)d_ws + 256);

    {   // zero p(0) and the barrier counter (deterministic per call)
        WaveFDTD2D_init_kernel<<<(Ncells + 255) / 256, 256, 0, stream>>>(buf, bar);
    }

    WaveFDTD2D_58926951301514_kernel<<<NB, TPB, 0, stream>>>(
        vel, src, src_x, src_z, rec_x, rec_z, out, buf, bar);
}